// SimpleGCN_71674414235956
// MI455X (gfx1250) — compile-verified
//
#include <hip/hip_runtime.h>

typedef float v2f __attribute__((ext_vector_type(2)));
typedef float v8f __attribute__((ext_vector_type(8)));

#define N_NODES 100000
#define N_EDGES 3200000
#define IN_CH   128
#define HID_CH  64
#define OUT_CH  32

// ---------------------------------------------------------------- utilities
__global__ __launch_bounds__(256) void zero_f32(float* __restrict__ p, int n) {
    int t = blockIdx.x * blockDim.x + threadIdx.x;
    if (t < n) p[t] = 0.0f;
}

__global__ __launch_bounds__(256) void degree_kernel(const int* __restrict__ dst,
                                                     const float* __restrict__ w,
                                                     float* __restrict__ deg) {
    int e = blockIdx.x * blockDim.x + threadIdx.x;
    if (e < N_EDGES) atomicAdd(&deg[dst[e]], w[e]);
}

__global__ __launch_bounds__(256) void dinv_kernel(float* __restrict__ deg_dinv) {
    int i = blockIdx.x * blockDim.x + threadIdx.x;
    if (i < N_NODES) {
        float d = deg_dinv[i] + 1.0f;   // self-loop weight 1
        deg_dinv[i] = rsqrtf(d);        // d >= 1 always
    }
}

// norm[e] = dinv[dst]*w*dinv[src], computed once, reused by both layers
__global__ __launch_bounds__(256) void norm_kernel(const int* __restrict__ src,
                                                   const int* __restrict__ dst,
                                                   const float* __restrict__ w,
                                                   const float* __restrict__ dinv,
                                                   float* __restrict__ norm) {
    int e = blockIdx.x * blockDim.x + threadIdx.x;
    if (e < N_EDGES) norm[e] = dinv[dst[e]] * w[e] * dinv[src[e]];
}

// ---------------------------------------------------------------- WMMA GEMM
// H[nrows x N] = X[nrows x K] @ W[K x N]   (fp32, V_WMMA_F32_16X16X4_F32)
// Block: 128 threads = 4 waves; each wave computes a 16-row x N stripe.
template <int K, int N>
__global__ __launch_bounds__(128) void gemm_wmma(const float* __restrict__ X,
                                                 const float* __restrict__ W,
                                                 float* __restrict__ H,
                                                 int nrows) {
    constexpr int NT = N / 16;                 // 16-wide column tiles per wave
    __shared__ float Ws[K * N];                // W1: 32KB, W2: 8KB (of 320KB LDS)

    // stage W into LDS (float4)
    for (int idx = threadIdx.x; idx < (K * N) / 4; idx += blockDim.x) {
        ((float4*)Ws)[idx] = ((const float4*)W)[idx];
    }
    __syncthreads();

    const int wave = threadIdx.x >> 5;
    const int lane = threadIdx.x & 31;
    const int half = lane >> 4;                // 0: lanes 0-15, 1: lanes 16-31
    const int l16  = lane & 15;
    const int rowBase = blockIdx.x * 64 + wave * 16;

    // clamp row for A loads (stores are guarded; EXEC stays all-ones for WMMA)
    int arow = rowBase + l16;
    if (arow >= nrows) arow = nrows - 1;
    const float* xr = X + (size_t)arow * K;

    v8f acc[NT] = {};                          // zero accumulators

#pragma unroll 4
    for (int kk = 0; kk < K; kk += 4) {
        // A fragment 16x4: lane half selects K pair {kk,kk+1} or {kk+2,kk+3}
        v2f a;
        {
            float2 av = *(const float2*)(xr + kk + 2 * half);
            a.x = av.x; a.y = av.y;
        }
#pragma unroll
        for (int t = 0; t < NT; t++) {
            // B fragment 4x16: column = l16, rows = kk+2*half, kk+2*half+1
            v2f b;
            b.x = Ws[(kk + 2 * half + 0) * N + t * 16 + l16];
            b.y = Ws[(kk + 2 * half + 1) * N + t * 16 + l16];
            acc[t] = __builtin_amdgcn_wmma_f32_16x16x4_f32(
                false, a, false, b, (short)0, acc[t], false, false);
        }
    }

    // D layout: VGPR r -> row r (lanes 0-15) / row r+8 (lanes 16-31), col = l16
    if (rowBase + 16 <= nrows) {
        // uniform fast path: whole 16-row tile in range, no per-lane guards
#pragma unroll
        for (int t = 0; t < NT; t++) {
#pragma unroll
            for (int r = 0; r < 8; r++) {
                int row = rowBase + r + 8 * half;
                H[(size_t)row * N + t * 16 + l16] = acc[t][r];
            }
        }
    } else {
#pragma unroll
        for (int t = 0; t < NT; t++) {
#pragma unroll
            for (int r = 0; r < 8; r++) {
                int row = rowBase + r + 8 * half;
                if (row < nrows) H[(size_t)row * N + t * 16 + l16] = acc[t][r];
            }
        }
    }
}

// ---------------------------------------------------------------- aggregation
// agg[dst] += norm[e] * h[src]; 16 threads per edge, each owns a contiguous
// CH/16 slice -> coalesced 128b/64b gathers, atomics spread over channels.
template <int CH>
__global__ __launch_bounds__(256) void aggregate_kernel(const int* __restrict__ src,
                                                        const int* __restrict__ dst,
                                                        const float* __restrict__ norm,
                                                        const float* __restrict__ h,
                                                        float* __restrict__ agg) {
    constexpr int PER = CH / 16;               // 4 (layer1) or 2 (layer2)
    int t = blockIdx.x * blockDim.x + threadIdx.x;   // < N_EDGES*16 = 51.2M
    int e    = t >> 4;
    int part = t & 15;
    if (e >= N_EDGES) return;

    int s = src[e], d = dst[e];
    float nm = norm[e];

    const float* hs = h + (size_t)s * CH + part * PER;
    float*       ad = agg + (size_t)d * CH + part * PER;
#pragma unroll
    for (int i = 0; i < PER; i++) {
        atomicAdd(ad + i, nm * hs[i]);
    }
}

// ---------------------------------------------------------------- finalize
// agg[i][c] += dinv[i]^2 * h[i][c] + bias[c]; optional ReLU (in place)
template <int CH, bool RELU>
__global__ __launch_bounds__(256) void finalize_kernel(const float* __restrict__ h,
                                                       const float* __restrict__ dinv,
                                                       const float* __restrict__ bias,
                                                       float* __restrict__ agg) {
    int t = blockIdx.x * blockDim.x + threadIdx.x;
    if (t >= N_NODES * CH) return;
    int i = t / CH;
    int c = t % CH;
    float di = dinv[i];
    float v = agg[t] + di * di * h[t] + bias[c];
    if (RELU) v = fmaxf(v, 0.0f);
    agg[t] = v;
}

// ---------------------------------------------------------------- launcher
extern "C" void kernel_launch(void* const* d_in, const int* in_sizes, int n_in,
                              void* d_out, int out_size, void* d_ws, size_t ws_size,
                              hipStream_t stream) {
    (void)in_sizes; (void)n_in; (void)out_size; (void)ws_size;

    const float* x   = (const float*)d_in[0];
    const int*   ei  = (const int*)d_in[1];     // [2, N_EDGES]
    const float* w   = (const float*)d_in[2];
    const float* W1  = (const float*)d_in[3];
    const float* b1  = (const float*)d_in[4];
    const float* W2  = (const float*)d_in[5];
    const float* b2  = (const float*)d_in[6];
    const int* src = ei;
    const int* dst = ei + N_EDGES;
    float* out = (float*)d_out;

    // workspace layout (all offsets 16B aligned):
    //   dinv : N_NODES floats                (doubles as deg accumulator)
    //   norm : N_EDGES floats                (per-edge coefficient, both layers)
    //   h1   : N_NODES*HID_CH floats         (reused as h2 after layer 1)
    //   agg1 : N_NODES*HID_CH floats         (reused as h_relu after finalize1)
    char* ws = (char*)d_ws;
    float* dinv = (float*)ws;
    float* norm = dinv + N_NODES;               // 100000*4 B -> 16B aligned
    float* h1   = norm + N_EDGES;
    float* agg1 = h1 + (size_t)N_NODES * HID_CH;
    float* h2   = h1;                           // alias: h1 dead after finalize1

    const int B = 256;
    // zero accumulators (d_out is poisoned by harness; we own re-zeroing it)
    zero_f32<<<(N_NODES + B - 1) / B, B, 0, stream>>>(dinv, N_NODES);
    zero_f32<<<(N_NODES * HID_CH + B - 1) / B, B, 0, stream>>>(agg1, N_NODES * HID_CH);
    zero_f32<<<(N_NODES * OUT_CH + B - 1) / B, B, 0, stream>>>(out, N_NODES * OUT_CH);

    // degrees -> dinv -> per-edge norm
    degree_kernel<<<(N_EDGES + B - 1) / B, B, 0, stream>>>(dst, w, dinv);
    dinv_kernel<<<(N_NODES + B - 1) / B, B, 0, stream>>>(dinv);
    norm_kernel<<<(N_EDGES + B - 1) / B, B, 0, stream>>>(src, dst, w, dinv, norm);

    // ---- layer 1 ----
    const int gemmBlocks = (N_NODES + 63) / 64;  // 64 rows per block (4 waves)
    gemm_wmma<IN_CH, HID_CH><<<gemmBlocks, 128, 0, stream>>>(x, W1, h1, N_NODES);
    {
        long long threads = (long long)N_EDGES * 16;
        int blocks = (int)((threads + B - 1) / B);
        aggregate_kernel<HID_CH><<<blocks, B, 0, stream>>>(src, dst, norm, h1, agg1);
    }
    finalize_kernel<HID_CH, true><<<(N_NODES * HID_CH + B - 1) / B, B, 0, stream>>>(
        h1, dinv, b1, agg1);                     // agg1 now holds relu(h)

    // ---- layer 2 ----
    gemm_wmma<HID_CH, OUT_CH><<<gemmBlocks, 128, 0, stream>>>(agg1, W2, h2, N_NODES);
    {
        long long threads = (long long)N_EDGES * 16;
        int blocks = (int)((threads + B - 1) / B);
        aggregate_kernel<OUT_CH><<<blocks, B, 0, stream>>>(src, dst, norm, h2, out);
    }
    finalize_kernel<OUT_CH, false><<<(N_NODES * OUT_CH + B - 1) / B, B, 0, stream>>>(
        h2, dinv, b2, out);
}